// ELG_TSP_Solver_35235911696918
// MI455X (gfx1250) — compile-verified
//
#include <hip/hip_runtime.h>
#include <hip/hip_bf16.h>

typedef float v2f __attribute__((ext_vector_type(2)));
typedef float v8f __attribute__((ext_vector_type(8)));

#define EDIM   128
#define HEADS  4
#define DH     32
#define NNODE  1000
#define BATCH  4
#define MROWS  (BATCH * NNODE)   // 4000, multiple of 16
#define KNNK   10
#define XIPEN  10.0f
#define INV_SQRT_DH 0.17677669529663687f   // 1/sqrt(32)
#define INV_SQRT_E  0.08838834764831845f   // 1/sqrt(128)

// ---------------------------------------------------------------------------
// C[M x N] = A[M x K] @ W[K x N] + bias, optional ReLU.  f32 WMMA 16x16x4.
// grid.x = M/16 tiles, grid.y * 8 waves cover N/16 tiles. block = 256 (8 waves)
// ---------------------------------------------------------------------------
__global__ __launch_bounds__(256) void wmma_gemm_nn(
    const float* __restrict__ A, const float* __restrict__ W,
    const float* __restrict__ bias, float* __restrict__ C,
    int K, int Nw, int relu) {
  const int lane  = threadIdx.x & 31;
  const int wave  = threadIdx.x >> 5;
  const int tileM = blockIdx.x;
  const int tileN = blockIdx.y * 8 + wave;
  if (tileN * 16 >= Nw) return;

  const int row   = tileM * 16 + (lane & 15);   // A row for this lane
  const int col   = tileN * 16 + (lane & 15);   // B/C column for this lane
  const int khalf = (lane >> 4) * 2;            // lanes 16-31 hold K+2,K+3

  v8f c = {};
  for (int k0 = 0; k0 < K; k0 += 4) {
    const int ks = k0 + khalf;
    v2f a, b;
    a.x = A[(size_t)row * K + ks];
    a.y = A[(size_t)row * K + ks + 1];
    b.x = W[(size_t)ks * Nw + col];
    b.y = W[(size_t)(ks + 1) * Nw + col];
    c = __builtin_amdgcn_wmma_f32_16x16x4_f32(false, a, false, b,
                                              (short)0, c, false, false);
  }
  const float bv = bias[col];
  const int mbase = tileM * 16 + ((lane >> 4) << 3);  // +8 for lanes 16-31
#pragma unroll
  for (int r = 0; r < 8; ++r) {
    float v = c[r] + bv;
    if (relu) v = fmaxf(v, 0.0f);
    C[(size_t)(mbase + r) * Nw + col] = v;
  }
}

// ---------------------------------------------------------------------------
// Per-batch: C[b] = A[b] (Mt x K) @ B[b]^T (K x Nt) * scale   (B row-major NtxK)
// grid = (ceil(Mt/16), ceil(ntiles/8), BATCH), block = 256
// ---------------------------------------------------------------------------
__global__ __launch_bounds__(256) void wmma_gemm_nt(
    const float* __restrict__ Aall, const float* __restrict__ Ball,
    float* __restrict__ Call, int Mt, int K, int Nt, float scale) {
  const int bz = blockIdx.z;
  const float* A = Aall + (size_t)bz * Mt * K;
  const float* B = Ball + (size_t)bz * Nt * K;
  float*       C = Call + (size_t)bz * Mt * Nt;

  const int lane  = threadIdx.x & 31;
  const int wave  = threadIdx.x >> 5;
  const int tileM = blockIdx.x;
  const int tileN = blockIdx.y * 8 + wave;
  const int ntile = (Nt + 15) / 16;
  if (tileN >= ntile) return;

  int row = tileM * 16 + (lane & 15); if (row >= Mt) row = Mt - 1;
  int col = tileN * 16 + (lane & 15); if (col >= Nt) col = Nt - 1;
  const int khalf = (lane >> 4) * 2;

  v8f c = {};
  for (int k0 = 0; k0 < K; k0 += 4) {
    const int ks = k0 + khalf;
    v2f a, b;
    a.x = A[(size_t)row * K + ks];
    a.y = A[(size_t)row * K + ks + 1];
    b.x = B[(size_t)col * K + ks];
    b.y = B[(size_t)col * K + ks + 1];
    c = __builtin_amdgcn_wmma_f32_16x16x4_f32(false, a, false, b,
                                              (short)0, c, false, false);
  }
  const int c0 = tileN * 16 + (lane & 15);
  if (c0 < Nt) {
    const int mbase = tileM * 16 + ((lane >> 4) << 3);
#pragma unroll
    for (int r = 0; r < 8; ++r) {
      const int m = mbase + r;
      if (m < Mt) C[(size_t)m * Nt + c0] = c[r] * scale;
    }
  }
}

// ---------------------------------------------------------------------------
// h = nodes @ in_w(2xE) + in_b
// ---------------------------------------------------------------------------
__global__ void embed_kernel(const float* __restrict__ nodes,
                             const float* __restrict__ in_w,
                             const float* __restrict__ in_b,
                             float* __restrict__ x) {
  const int i = blockIdx.x * 256 + threadIdx.x;
  if (i >= MROWS * EDIM) return;
  const int row = i >> 7, e = i & 127;
  x[i] = nodes[row * 2] * in_w[e] + nodes[row * 2 + 1] * in_w[EDIM + e] + in_b[e];
}

// ---------------------------------------------------------------------------
// Softmax attention for one (batch, head), 8 query rows / block.
// qkv rows are 3E=384 wide: q [0:128), k [128:256), v [256:384).
// ---------------------------------------------------------------------------
__global__ __launch_bounds__(256) void attn_kernel(const float* __restrict__ qkv,
                                                   float* __restrict__ O) {
  const int tid = threadIdx.x;
  const int bh  = blockIdx.y;
  const int b   = bh >> 2;
  const int hd  = bh & 3;
  const int row0 = blockIdx.x * 8;
  const size_t base = (size_t)b * NNODE * 384;

  __shared__ float sQ[8][DH];
  __shared__ float sS[8][NNODE];
  __shared__ float sP[8][32];
  __shared__ float sMax[8], sSum[8];

  { const int r = tid >> 5, d = tid & 31;
    sQ[r][d] = qkv[base + (size_t)(row0 + r) * 384 + hd * DH + d]; }
  __syncthreads();

  for (int i = tid; i < 8 * NNODE; i += 256) {
    const int r = i & 7, k = i >> 3;
    const float* kp = qkv + base + (size_t)k * 384 + EDIM + hd * DH;
    float acc = 0.f;
#pragma unroll
    for (int d = 0; d < DH; ++d) acc += sQ[r][d] * kp[d];
    sS[r][k] = acc * INV_SQRT_DH;
  }
  __syncthreads();

  const int r = tid >> 5, l = tid & 31;
  float m = -1e38f;
  for (int k = l; k < NNODE; k += 32) m = fmaxf(m, sS[r][k]);
  sP[r][l] = m; __syncthreads();
  if (l == 0) { float mm = -1e38f; for (int j = 0; j < 32; ++j) mm = fmaxf(mm, sP[r][j]); sMax[r] = mm; }
  __syncthreads();
  const float rm = sMax[r];
  float s = 0.f;
  for (int k = l; k < NNODE; k += 32) { float e_ = expf(sS[r][k] - rm); sS[r][k] = e_; s += e_; }
  sP[r][l] = s; __syncthreads();
  if (l == 0) { float ss = 0.f; for (int j = 0; j < 32; ++j) ss += sP[r][j]; sSum[r] = ss; }
  __syncthreads();
  const float inv = 1.f / sSum[r];

  float acc = 0.f;
  for (int k = 0; k < NNODE; ++k)
    acc += sS[r][k] * qkv[base + (size_t)k * 384 + 2 * EDIM + hd * DH + l];
  O[((size_t)b * NNODE + row0 + r) * EDIM + hd * DH + l] = acc * inv;
}

// ---------------------------------------------------------------------------
// x = LayerNorm(x + t) * g + b ; one 128-elem row per 128-thread block
// ---------------------------------------------------------------------------
__global__ void residual_ln_kernel(float* __restrict__ x, const float* __restrict__ t,
                                   const float* __restrict__ g, const float* __restrict__ bt) {
  const int row = blockIdx.x, e = threadIdx.x;
  __shared__ float s[EDIM];
  __shared__ float s_mean, s_var;
  const float v = x[(size_t)row * EDIM + e] + t[(size_t)row * EDIM + e];
  s[e] = v; __syncthreads();
  for (int off = 64; off > 0; off >>= 1) { if (e < off) s[e] += s[e + off]; __syncthreads(); }
  if (e == 0) s_mean = s[0] * (1.f / EDIM);
  __syncthreads();
  const float d = v - s_mean;
  s[e] = d * d; __syncthreads();
  for (int off = 64; off > 0; off >>= 1) { if (e < off) s[e] += s[e + off]; __syncthreads(); }
  if (e == 0) s_var = s[0] * (1.f / EDIM);
  __syncthreads();
  x[(size_t)row * EDIM + e] = d * rsqrtf(s_var + 1e-5f) * g[e] + bt[e];
}

// ---------------------------------------------------------------------------
// ql = ctx @ lq_w + lq_b  (step-invariant, hoisted out of decode loop)
// ---------------------------------------------------------------------------
__global__ void ql_kernel(const float* __restrict__ ctx, const float* __restrict__ lq_w,
                          const float* __restrict__ lq_b, float* __restrict__ qlv) {
  const int e = threadIdx.x;
  float acc = lq_b[e];
  for (int j = 0; j < EDIM; ++j) acc += ctx[j] * lq_w[j * EDIM + e];
  qlv[e] = acc;
}

// ---------------------------------------------------------------------------
// Sequential greedy decode: one block per batch, 999 steps in-kernel.
// ---------------------------------------------------------------------------
__global__ __launch_bounds__(256) void decode_kernel(
    const float* __restrict__ nodes, const float* __restrict__ ug,
    const float* __restrict__ qlv,
    const float* __restrict__ l_in_w, const float* __restrict__ l_in_b,
    const float* __restrict__ lk_w, const float* __restrict__ lk_b,
    const float* __restrict__ lv_w, const float* __restrict__ lv_b,
    const float* __restrict__ lo_w, const float* __restrict__ lo_b,
    const int* __restrict__ start_idx_p, int* __restrict__ tour) {
  const int b = blockIdx.x, tid = threadIdx.x;

  __shared__ float s_nodes[2 * NNODE];
  __shared__ float s_dm[NNODE];
  __shared__ unsigned char s_vis[NNODE];
  __shared__ float s_le[KNNK][EDIM], s_kk[KNNK][EDIM], s_vv[KNNK][EDIM];
  __shared__ float s_red[256]; __shared__ int s_redi[256];
  __shared__ float s_ao[EDIM], s_ao2[EDIM], s_ql[EDIM];
  __shared__ float s_al[HEADS][KNNK];
  __shared__ float s_sl[KNNK];
  __shared__ int   s_idx[KNNK];
  __shared__ float s_rho[KNNK], s_theta[KNNK];
  __shared__ int   s_curr;
  __shared__ float s_rhomax;

  for (int i = tid; i < 2 * NNODE; i += 256) s_nodes[i] = nodes[(size_t)b * 2 * NNODE + i];
  for (int i = tid; i < NNODE; i += 256) s_vis[i] = 0;
  if (tid < EDIM) s_ql[tid] = qlv[tid];
  __syncthreads();
  if (tid == 0) {
    const int s0 = *start_idx_p;
    s_vis[s0] = 1; s_curr = s0; tour[(size_t)b * NNODE] = s0;
  }
  __syncthreads();

  for (int step = 0; step < NNODE - 1; ++step) {
    const int curr = s_curr;
    const float ccx = s_nodes[2 * curr], ccy = s_nodes[2 * curr + 1];

    // masked distances
    for (int n = tid; n < NNODE; n += 256) {
      const float dx = s_nodes[2 * n] - ccx, dy = s_nodes[2 * n + 1] - ccy;
      s_dm[n] = s_vis[n] ? 1e30f : sqrtf(dx * dx + dy * dy);
    }
    __syncthreads();

    // top-KNN smallest (stable: lowest index wins ties, matching lax.top_k)
    for (int kk = 0; kk < KNNK; ++kk) {
      float bv = 1e38f; int bi = NNODE;
      for (int n = tid; n < NNODE; n += 256) {
        const float v = s_dm[n];
        if (v < bv || (v == bv && n < bi)) { bv = v; bi = n; }
      }
      s_red[tid] = bv; s_redi[tid] = bi; __syncthreads();
      for (int off = 128; off > 0; off >>= 1) {
        if (tid < off) {
          const float ov = s_red[tid + off]; const int oi = s_redi[tid + off];
          if (ov < s_red[tid] || (ov == s_red[tid] && oi < s_redi[tid])) { s_red[tid] = ov; s_redi[tid] = oi; }
        }
        __syncthreads();
      }
      if (tid == 0) { s_idx[kk] = s_redi[0]; s_dm[s_redi[0]] = 1e38f; }
      __syncthreads();
    }

    // polar features
    if (tid < KNNK) {
      const int id = s_idx[tid];
      const float dx = s_nodes[2 * id] - ccx, dy = s_nodes[2 * id + 1] - ccy;
      s_rho[tid] = sqrtf(dx * dx + dy * dy + 1e-8f);
      s_theta[tid] = atan2f(dy, dx);
    }
    __syncthreads();
    if (tid == 0) { float mx = 0.f; for (int k = 0; k < KNNK; ++k) mx = fmaxf(mx, s_rho[k]); s_rhomax = mx + 1e-6f; }
    __syncthreads();

    // le = polar @ l_in_w + l_in_b
    for (int i = tid; i < KNNK * EDIM; i += 256) {
      const int k = i / EDIM, e = i % EDIM;
      s_le[k][e] = (s_rho[k] / s_rhomax) * l_in_w[e] + s_theta[k] * l_in_w[EDIM + e] + l_in_b[e];
    }
    __syncthreads();

    // kk = le@lk_w+b ; vv = le@lv_w+b
    for (int i = tid; i < KNNK * EDIM; i += 256) {
      const int k = i / EDIM, e = i % EDIM;
      float a1 = lk_b[e], a2 = lv_b[e];
      for (int j = 0; j < EDIM; ++j) {
        const float lv = s_le[k][j];
        a1 += lv * lk_w[j * EDIM + e];
        a2 += lv * lv_w[j * EDIM + e];
      }
      s_kk[k][e] = a1; s_vv[k][e] = a2;
    }
    __syncthreads();

    // local attention logits + softmax per head
    if (tid < HEADS * KNNK) {
      const int h = tid / KNNK, k = tid % KNNK;
      float acc = 0.f;
      for (int d = 0; d < DH; ++d) acc += s_ql[h * DH + d] * s_kk[k][h * DH + d];
      s_al[h][k] = acc * INV_SQRT_DH;
    }
    __syncthreads();
    if (tid < HEADS) {
      float mx = -1e38f;
      for (int k = 0; k < KNNK; ++k) mx = fmaxf(mx, s_al[tid][k]);
      float ss = 0.f;
      for (int k = 0; k < KNNK; ++k) { const float e_ = expf(s_al[tid][k] - mx); s_al[tid][k] = e_; ss += e_; }
      const float inv = 1.f / ss;
      for (int k = 0; k < KNNK; ++k) s_al[tid][k] *= inv;
    }
    __syncthreads();

    if (tid < EDIM) {
      const int h = tid / DH;
      float acc = 0.f;
      for (int k = 0; k < KNNK; ++k) acc += s_al[h][k] * s_vv[k][tid];
      s_ao[tid] = acc;
    }
    __syncthreads();
    if (tid < EDIM) {
      float acc = lo_b[tid];
      for (int j = 0; j < EDIM; ++j) acc += s_ao[j] * lo_w[j * EDIM + tid];
      s_ao2[tid] = acc;
    }
    __syncthreads();
    if (tid < KNNK) {
      float acc = 0.f;
      for (int j = 0; j < EDIM; ++j) acc += s_ao2[j] * s_le[tid][j];
      s_sl[tid] = acc * INV_SQRT_E;
    }
    __syncthreads();

    // u = ug - XI*penalty + u_l ; argmax (first-max tie rule)
    const float* ugrow = ug + ((size_t)b * NNODE + curr) * NNODE;
    float bv = -1e38f; int bi = NNODE;
    for (int n = tid; n < NNODE; n += 256) {
      float u;
      if (s_vis[n]) u = -1e38f;
      else {
        float pen = XIPEN, ul = 0.f;
        for (int k = 0; k < KNNK; ++k) if (s_idx[k] == n) { pen = 0.f; ul = s_sl[k]; }
        u = ugrow[n] - pen + ul;
      }
      if (u > bv || (u == bv && n < bi)) { bv = u; bi = n; }
    }
    s_red[tid] = bv; s_redi[tid] = bi; __syncthreads();
    for (int off = 128; off > 0; off >>= 1) {
      if (tid < off) {
        const float ov = s_red[tid + off]; const int oi = s_redi[tid + off];
        if (ov > s_red[tid] || (ov == s_red[tid] && oi < s_redi[tid])) { s_red[tid] = ov; s_redi[tid] = oi; }
      }
      __syncthreads();
    }
    if (tid == 0) {
      const int nxt = s_redi[0];
      s_vis[nxt] = 1; s_curr = nxt;
      tour[(size_t)b * NNODE + step + 1] = nxt;
    }
    __syncthreads();
  }
}

// ---------------------------------------------------------------------------
// Host side. Input order assumes JAX pytree flattening (dict keys sorted):
//  0 nodes | 1 ctx | 2 in_b | 3 in_w | 4 k_b | 5 k_w | 6 l_in_b | 7 l_in_w |
//  8+12L: f1_b,f1_w,f2_b,f2_w,ln1_b,ln1_g,ln2_b,ln2_g,o_b,o_w,qkv_b,qkv_w |
//  32 lk_b | 33 lk_w | 34 lo_b | 35 lo_w | 36 lq_b | 37 lq_w | 38 lv_b |
//  39 lv_w | 40 q_b | 41 q_w | 42 start_idx
// ---------------------------------------------------------------------------
extern "C" void kernel_launch(void* const* d_in, const int* in_sizes, int n_in,
                              void* d_out, int out_size, void* d_ws, size_t ws_size,
                              hipStream_t stream) {
  const float* nodes  = (const float*)d_in[0];
  const float* ctx    = (const float*)d_in[1];
  const float* in_b   = (const float*)d_in[2];
  const float* in_w   = (const float*)d_in[3];
  const float* k_b    = (const float*)d_in[4];
  const float* k_w    = (const float*)d_in[5];
  const float* l_in_b = (const float*)d_in[6];
  const float* l_in_w = (const float*)d_in[7];
  const float* lk_b   = (const float*)d_in[32];
  const float* lk_w   = (const float*)d_in[33];
  const float* lo_b   = (const float*)d_in[34];
  const float* lo_w   = (const float*)d_in[35];
  const float* lq_b   = (const float*)d_in[36];
  const float* lq_w   = (const float*)d_in[37];
  const float* lv_b   = (const float*)d_in[38];
  const float* lv_w   = (const float*)d_in[39];
  const float* q_b    = (const float*)d_in[40];
  const float* q_w    = (const float*)d_in[41];
  const int*   start  = (const int*)d_in[42];
  int* tour = (int*)d_out;

  float* ws = (float*)d_ws;
  size_t off = 0;
  float* x     = ws + off; off += (size_t)MROWS * EDIM;        // 512000
  float* tmp   = ws + off; off += (size_t)MROWS * 3 * EDIM;    // 1536000 (qkv / f1+f2)
  float* tmp2  = tmp + (size_t)MROWS * EDIM;                   // second third of tmp
  float* attnO = ws + off; off += (size_t)MROWS * EDIM;        // 512000
  float* qall  = ws + off; off += (size_t)MROWS * EDIM;        // 512000
  float* kmat  = ws + off; off += (size_t)MROWS * EDIM;        // 512000
  float* ugb   = ws + off; off += (size_t)BATCH * NNODE * NNODE; // 4000000
  float* qlv   = ws + off; off += EDIM;

  embed_kernel<<<dim3((MROWS * EDIM + 255) / 256), 256, 0, stream>>>(nodes, in_w, in_b, x);

  for (int l = 0; l < 2; ++l) {
    const void* const* p = d_in + 8 + 12 * l;
    const float* f1_b  = (const float*)p[0];
    const float* f1_w  = (const float*)p[1];
    const float* f2_b  = (const float*)p[2];
    const float* f2_w  = (const float*)p[3];
    const float* ln1_b = (const float*)p[4];
    const float* ln1_g = (const float*)p[5];
    const float* ln2_b = (const float*)p[6];
    const float* ln2_g = (const float*)p[7];
    const float* o_b   = (const float*)p[8];
    const float* o_w   = (const float*)p[9];
    const float* qkv_b = (const float*)p[10];
    const float* qkv_w = (const float*)p[11];

    wmma_gemm_nn<<<dim3(MROWS / 16, 3), 256, 0, stream>>>(x, qkv_w, qkv_b, tmp, EDIM, 3 * EDIM, 0);
    attn_kernel<<<dim3(NNODE / 8, BATCH * HEADS), 256, 0, stream>>>(tmp, attnO);
    wmma_gemm_nn<<<dim3(MROWS / 16, 1), 256, 0, stream>>>(attnO, o_w, o_b, tmp, EDIM, EDIM, 0);
    residual_ln_kernel<<<MROWS, EDIM, 0, stream>>>(x, tmp, ln1_g, ln1_b);
    wmma_gemm_nn<<<dim3(MROWS / 16, 1), 256, 0, stream>>>(x, f1_w, f1_b, tmp, EDIM, EDIM, 1);
    wmma_gemm_nn<<<dim3(MROWS / 16, 1), 256, 0, stream>>>(tmp, f2_w, f2_b, tmp2, EDIM, EDIM, 0);
    residual_ln_kernel<<<MROWS, EDIM, 0, stream>>>(x, tmp2, ln2_g, ln2_b);
  }

  // hoisted decode precompute: Qall, Kmat, UG = Qall @ Kmat^T / sqrt(E), ql
  wmma_gemm_nn<<<dim3(MROWS / 16, 1), 256, 0, stream>>>(x, q_w, q_b, qall, EDIM, EDIM, 0);
  wmma_gemm_nn<<<dim3(MROWS / 16, 1), 256, 0, stream>>>(x, k_w, k_b, kmat, EDIM, EDIM, 0);
  wmma_gemm_nt<<<dim3(63, 8, BATCH), 256, 0, stream>>>(qall, kmat, ugb, NNODE, EDIM, NNODE, INV_SQRT_E);
  ql_kernel<<<1, EDIM, 0, stream>>>(ctx, lq_w, lq_b, qlv);

  decode_kernel<<<BATCH, 256, 0, stream>>>(nodes, ugb, qlv,
                                           l_in_w, l_in_b, lk_w, lk_b, lv_w, lv_b,
                                           lo_w, lo_b, start, tour);
}